// GreedyTokenSelector_36567351558902
// MI455X (gfx1250) — compile-verified
//
#include <hip/hip_runtime.h>
#include <hip/hip_bf16.h>

// Problem constants (from reference)
#define H 16
#define NTOK 4096
#define DDIM 1024
#define THRESHOLD 0.01f
#define NEG_INF -1000000000.0f

typedef __attribute__((ext_vector_type(2))) float v2f;
typedef __attribute__((ext_vector_type(8))) float v8f;

// ---------------------------------------------------------------------------
// Kernel 0: zero the key mask (d_ws is poisoned by the harness)
// ---------------------------------------------------------------------------
__global__ void gts_zero_mask(int* __restrict__ mask) {
    int t = blockIdx.x * blockDim.x + threadIdx.x;
    if (t < NTOK) mask[t] = 0;
}

// ---------------------------------------------------------------------------
// Kernel 1: per-row head-mean via WMMA + max/argmax + key-mask scatter
// One block (128 threads = 4 waves) per query row i.
// Each wave processes 16-column tiles: D = ones(16x4) x B(4x16) + C chained
// over 4 K-chunks sums all 16 heads in f32. c[0] at lane L holds the column
// sum for column (L & 15) (rows M=0 and M=8 of D are identical since A==1).
// ---------------------------------------------------------------------------
__global__ void gts_rowtop_kernel(const float* __restrict__ attn,
                                  int* __restrict__ mask) {
    const int i    = blockIdx.x;          // query row
    const int lane = threadIdx.x & 31;
    const int wave = threadIdx.x >> 5;    // 0..3
    const int half = lane >> 4;           // 0 or 1
    const int col  = lane & 15;           // column within 16-wide tile

    const float* rowBase = attn + (size_t)i * NTOK;
    const size_t planeStride = (size_t)NTOK * NTOK;

    v2f a; a.x = 1.0f; a.y = 1.0f;        // all-ones A matrix (16x4)

    float bestVal = NEG_INF;
    int   bestIdx = 0x7FFFFFFF;

    // 256 tiles of 16 columns; 4 waves stride through them
    for (int t = wave; t < NTOK / 16; t += 4) {
        const int j0 = t * 16;
        v8f c = {};
#pragma unroll
        for (int kc = 0; kc < 4; ++kc) {
            const int h0 = 4 * kc + 2 * half;   // which two head planes this half-wave loads
            v2f b;
            b.x = rowBase[(size_t)h0 * planeStride + j0 + col];
            b.y = rowBase[(size_t)(h0 + 1) * planeStride + j0 + col];
            // D = A(ones) x B + C  -> column sums accumulate over 16 heads
            c = __builtin_amdgcn_wmma_f32_16x16x4_f32(
                    /*neg_a=*/false, a, /*neg_b=*/false, b,
                    /*c_mod=*/(short)0, c, /*reuse_a=*/false, /*reuse_b=*/false);
        }
        const float mean  = c[0] * (1.0f / 16.0f);
        const float score = (mean < THRESHOLD) ? NEG_INF : mean;
        const int   jidx  = j0 + col;
        // first-occurrence argmax: strictly greater, or equal with smaller idx
        if (score > bestVal || (score == bestVal && jidx < bestIdx)) {
            bestVal = score;
            bestIdx = jidx;
        }
    }

    __shared__ float sVal[128];
    __shared__ int   sIdx[128];
    sVal[threadIdx.x] = bestVal;
    sIdx[threadIdx.x] = bestIdx;
    __syncthreads();

    if (threadIdx.x == 0) {
        float v = sVal[0];
        int   ix = sIdx[0];
        for (int t = 1; t < 128; ++t) {
            const float tv = sVal[t];
            const int   ti = sIdx[t];
            if (tv > v || (tv == v && ti < ix)) { v = tv; ix = ti; }
        }
        // valid iff top1_val >= threshold; racy stores of 1 are idempotent
        if (!(v < THRESHOLD)) mask[ix] = 1;
    }
}

// ---------------------------------------------------------------------------
// Kernel 2: out[i, :] = x[i, :] * key_mask[i]   (float4 vectorized)
// ---------------------------------------------------------------------------
__global__ void gts_scale_kernel(const float* __restrict__ x,
                                 const int* __restrict__ mask,
                                 float* __restrict__ out) {
    const size_t e    = ((size_t)blockIdx.x * blockDim.x + threadIdx.x) * 4;
    const int    row  = (int)(e >> 10);          // DDIM = 1024
    const float  m    = mask[row] ? 1.0f : 0.0f;
    const float4 xin  = *(const float4*)(x + e);
    float4 o;
    o.x = xin.x * m;
    o.y = xin.y * m;
    o.z = xin.z * m;
    o.w = xin.w * m;
    *(float4*)(out + e) = o;
}

// ---------------------------------------------------------------------------
// Launch
// ---------------------------------------------------------------------------
extern "C" void kernel_launch(void* const* d_in, const int* in_sizes, int n_in,
                              void* d_out, int out_size, void* d_ws, size_t ws_size,
                              hipStream_t stream) {
    const float* x    = (const float*)d_in[0];   // [NTOK, DDIM] f32
    const float* attn = (const float*)d_in[1];   // [H, NTOK, NTOK] f32
    float*       out  = (float*)d_out;           // [NTOK, DDIM] f32
    int*         mask = (int*)d_ws;              // NTOK ints of scratch

    gts_zero_mask<<<(NTOK + 255) / 256, 256, 0, stream>>>(mask);
    gts_rowtop_kernel<<<NTOK, 128, 0, stream>>>(attn, mask);
    gts_scale_kernel<<<(NTOK * (DDIM / 4) + 255) / 256, 256, 0, stream>>>(x, mask, out);
}